// ClfCapsule_28913719837231
// MI455X (gfx1250) — compile-verified
//
#include <hip/hip_runtime.h>
#include <math.h>

typedef __attribute__((ext_vector_type(2))) float v2f;
typedef __attribute__((ext_vector_type(8))) float v8f;

#define B_SZ   256
#define M_SZ   1024
#define A_SZ   8
#define K_CLS  20
#define O_SZ   16
#define KPAD   32
#define I_SZ   (B_SZ * A_SZ)   /* 2048 flattened (batch, atom) rows */

// ---------------------------------------------------------------------------
// Zero-init routing logits b_ij [M_SZ][KPAD]
// ---------------------------------------------------------------------------
__global__ void caps_zero_bij(float* __restrict__ bij) {
    int idx = blockIdx.x * blockDim.x + threadIdx.x;   // grid covers M_SZ*KPAD
    bij[idx] = 0.0f;
}

// ---------------------------------------------------------------------------
// Column softmax over m (1024) for each class k (20). b_ij, c are [M][KPAD].
// One block per class, 256 threads, 4 rows per thread.
// ---------------------------------------------------------------------------
__global__ void caps_softmax(const float* __restrict__ bij, float* __restrict__ c) {
    __shared__ float red[256];
    const int k = blockIdx.x;
    const int t = threadIdx.x;

    float v0 = bij[(t + 0  ) * KPAD + k];
    float v1 = bij[(t + 256) * KPAD + k];
    float v2 = bij[(t + 512) * KPAD + k];
    float v3 = bij[(t + 768) * KPAD + k];

    float mx = fmaxf(fmaxf(v0, v1), fmaxf(v2, v3));
    red[t] = mx;
    __syncthreads();
    for (int s = 128; s > 0; s >>= 1) {
        if (t < s) red[t] = fmaxf(red[t], red[t + s]);
        __syncthreads();
    }
    mx = red[0];
    __syncthreads();

    float e0 = expf(v0 - mx), e1 = expf(v1 - mx);
    float e2 = expf(v2 - mx), e3 = expf(v3 - mx);
    red[t] = e0 + e1 + e2 + e3;
    __syncthreads();
    for (int s = 128; s > 0; s >>= 1) {
        if (t < s) red[t] += red[t + s];
        __syncthreads();
    }
    float inv = 1.0f / red[0];

    c[(t + 0  ) * KPAD + k] = e0 * inv;
    c[(t + 256) * KPAD + k] = e1 * inv;
    c[(t + 512) * KPAD + k] = e2 * inv;
    c[(t + 768) * KPAD + k] = e3 * inv;
}

// ---------------------------------------------------------------------------
// GEMM 1 (fp32 WMMA): t2[i=(b*8+a)][k] = sum_m xt[b,m,a] * c[m][k]
//   A[i, m] = x_flat[b*8192 + m*8 + a]   (2048 x 1024)
//   B[m, k] = c[m*KPAD + k]              (1024 x 32, cols >=20 garbage/ignored)
// One wave computes a full 16x32 output slab (both k-tiles), sharing the
// A-tile loads between the two WMMAs. 128 waves -> 16 blocks x 8 waves.
// ---------------------------------------------------------------------------
__global__ void caps_gemm_t(const float* __restrict__ x,
                            const float* __restrict__ c,
                            float* __restrict__ t2) {
    const int wave = (blockIdx.x * blockDim.x + threadIdx.x) >> 5;
    const int lane = threadIdx.x & 31;
    const int i0   = wave * 16;
    const int lo   = lane & 15;
    const int hi   = lane >> 4;
    const int row  = i0 + lo;                               // A-matrix row
    const int xb   = (row >> 3) * (A_SZ * M_SZ) + (row & 7);  // + m*8
    const int j0   = hi * 2;                                // contraction sub-idx

    v8f acc0 = {};
    v8f acc1 = {};
    #pragma unroll 2
    for (int mb = 0; mb < M_SZ; mb += 4) {
        v2f av, bv0, bv1;
        av.x  = x[xb + (mb + j0    ) * A_SZ];
        av.y  = x[xb + (mb + j0 + 1) * A_SZ];
        bv0.x = c[(mb + j0    ) * KPAD + lo];
        bv0.y = c[(mb + j0 + 1) * KPAD + lo];
        bv1.x = c[(mb + j0    ) * KPAD + 16 + lo];
        bv1.y = c[(mb + j0 + 1) * KPAD + 16 + lo];
        acc0 = __builtin_amdgcn_wmma_f32_16x16x4_f32(
            false, av, false, bv0, (short)0, acc0, false, false);
        acc1 = __builtin_amdgcn_wmma_f32_16x16x4_f32(
            false, av, false, bv1, (short)0, acc1, false, false);
    }
    #pragma unroll
    for (int r = 0; r < 8; ++r) {
        const int rbase = (i0 + r + hi * 8) * KPAD;
        t2[rbase + lo]      = acc0[r];
        t2[rbase + 16 + lo] = acc1[r];
    }
}

// ---------------------------------------------------------------------------
// Squash + agreement pre-projection. One block per batch element, 320 threads.
//   s[k,o]  = sum_a W[k,o,a] * t2[(b*8+a)][k]
//   magsq[o]= sum_k s[k,o]^2   (squash over CLASS axis, faithful to reference)
//   v[k,o]  = s * magsq / ((1+magsq) * sqrt(magsq))      -> written to d_out
//   wv2[(b*8+a)][k] = (1/B) * sum_o W[k,o,a] * v[k,o]    (mean folded in here)
// ---------------------------------------------------------------------------
__global__ void caps_squash(const float* __restrict__ t2,
                            const float* __restrict__ W,
                            float* __restrict__ out,
                            float* __restrict__ wv2) {
    __shared__ float s_s[K_CLS][O_SZ];
    __shared__ float s_v[K_CLS][O_SZ];
    const int b = blockIdx.x;
    const int t = threadIdx.x;            // 0..319
    const int k = t >> 4;
    const int o = t & 15;

    float s = 0.0f;
    #pragma unroll
    for (int a = 0; a < A_SZ; ++a)
        s += W[k * (O_SZ * A_SZ) + o * A_SZ + a] * t2[(b * 8 + a) * KPAD + k];
    s_s[k][o] = s;
    __syncthreads();

    float magsq = 0.0f;
    #pragma unroll
    for (int kk = 0; kk < K_CLS; ++kk) {
        float sv = s_s[kk][o];
        magsq += sv * sv;
    }
    float scale = magsq / ((1.0f + magsq) * sqrtf(magsq));
    float v = s * scale;
    s_v[k][o] = v;
    out[b * (K_CLS * O_SZ) + t] = v;      // t == k*16 + o
    __syncthreads();

    if (t < K_CLS * A_SZ) {               // 160 threads
        int k2 = t % K_CLS;
        int a  = t / K_CLS;
        float w = 0.0f;
        #pragma unroll
        for (int o2 = 0; o2 < O_SZ; ++o2)
            w += W[k2 * (O_SZ * A_SZ) + o2 * A_SZ + a] * s_v[k2][o2];
        wv2[(b * 8 + a) * KPAD + k2] = w * (1.0f / (float)B_SZ);
    }
}

// ---------------------------------------------------------------------------
// GEMM 2 (fp32 WMMA): bij[m][k] += sum_i xt[b,m,a] * wv2[i][k]
//   A[m, i] = x_flat[(i>>3)*8192 + m*8 + (i&7)]   (1024 x 2048)
//   B[i, k] = wv2[i*KPAD + k]                     (2048 x 32)
// One wave per 16x32 output slab; 64 waves -> 8 blocks x 8 waves.
// ---------------------------------------------------------------------------
__global__ void caps_gemm_b(const float* __restrict__ x,
                            const float* __restrict__ wv2,
                            float* __restrict__ bij) {
    const int wave = (blockIdx.x * blockDim.x + threadIdx.x) >> 5;
    const int lane = threadIdx.x & 31;
    const int m0   = wave * 16;
    const int lo   = lane & 15;
    const int hi   = lane >> 4;
    const int m    = m0 + lo;
    const int j0   = hi * 2;

    v8f acc0 = {};
    v8f acc1 = {};
    #pragma unroll 2
    for (int ib = 0; ib < I_SZ; ib += 4) {
        const int ia = ib + j0;
        const int ic = ia + 1;
        v2f av, bv0, bv1;
        av.x  = x[(ia >> 3) * (A_SZ * M_SZ) + m * A_SZ + (ia & 7)];
        av.y  = x[(ic >> 3) * (A_SZ * M_SZ) + m * A_SZ + (ic & 7)];
        bv0.x = wv2[ia * KPAD + lo];
        bv0.y = wv2[ic * KPAD + lo];
        bv1.x = wv2[ia * KPAD + 16 + lo];
        bv1.y = wv2[ic * KPAD + 16 + lo];
        acc0 = __builtin_amdgcn_wmma_f32_16x16x4_f32(
            false, av, false, bv0, (short)0, acc0, false, false);
        acc1 = __builtin_amdgcn_wmma_f32_16x16x4_f32(
            false, av, false, bv1, (short)0, acc1, false, false);
    }
    #pragma unroll
    for (int r = 0; r < 8; ++r) {
        const int rbase = (m0 + r + hi * 8) * KPAD;
        bij[rbase + lo]      += acc0[r];
        bij[rbase + 16 + lo] += acc1[r];
    }
}

// ---------------------------------------------------------------------------
extern "C" void kernel_launch(void* const* d_in, const int* in_sizes, int n_in,
                              void* d_out, int out_size, void* d_ws, size_t ws_size,
                              hipStream_t stream) {
    (void)in_sizes; (void)n_in; (void)out_size; (void)ws_size;
    const float* x = (const float*)d_in[0];   // [256, 8, 1024] == xt view [256,1024,8]
    const float* W = (const float*)d_in[1];   // [20, 16, 8]
    float* out = (float*)d_out;               // [256, 20, 16, 1]

    float* bij = (float*)d_ws;                // [1024][32]  128 KB
    float* c   = bij + M_SZ * KPAD;           // [1024][32]  128 KB
    float* t2  = c   + M_SZ * KPAD;           // [2048][32]  256 KB
    float* wv2 = t2  + I_SZ * KPAD;           // [2048][32]  256 KB

    caps_zero_bij<<<(M_SZ * KPAD) / 256, 256, 0, stream>>>(bij);

    for (int it = 0; it < 3; ++it) {
        caps_softmax<<<K_CLS, 256, 0, stream>>>(bij, c);
        caps_gemm_t<<<16, 256, 0, stream>>>(x, c, t2);
        caps_squash<<<B_SZ, 320, 0, stream>>>(t2, W, out, wv2);
        if (it < 2)
            caps_gemm_b<<<8, 256, 0, stream>>>(x, wv2, bij);
    }
}